// GeneralizedMultichannelWienerFiter_26886495273326
// MI455X (gfx1250) — compile-verified
//
#include <hip/hip_runtime.h>
#include <stdint.h>

// Generalized multichannel Wiener filter, B=8 T=250 F=257 M=8.
// A = s s^H + n n^H + eps I. Since rhs b = s*conj(s0) lies exactly in range(U), U=[s,n],
// and A * U * G^-1 = U with G = eps I2 + U^H U (2x2 Hermitian), the exact solution is
//   w  = U * G^-1 * [conj(s0), 0]^T
//   bf = w^H x = s0 * (g22 * (s^H x) - (s^H n) * (n^H x)) / det(G)
// -> 5 length-8 complex dots + a tiny fp64 2x2 tail. No 1/eps amplification anywhere.
// Memory bound: ~103 MB traffic -> ~4.4 us at 23.3 TB/s; VALU ~160 instr/thread (well under roof).

#define WF_EPS 1e-5f

static constexpr int kM    = 8;
static constexpr int kBins = 8 * 250 * 257;   // 514000
static constexpr int TPB   = 256;
static constexpr int kBinStrideLds = 20;      // 16 data floats + 4 pad per bin (80B, 16B-aligned, conflict-free-ish)

// ---- async global->LDS path availability (CDNA5 ASYNCcnt data mover) ----
#if defined(__has_builtin)
#  if __has_builtin(__builtin_amdgcn_global_load_async_to_lds_b128) && \
      __has_builtin(__builtin_amdgcn_s_wait_asynccnt)
#    define WF_ASYNC_LDS 1
#  endif
#endif
#ifndef WF_ASYNC_LDS
#  define WF_ASYNC_LDS 0
#endif

#if WF_ASYNC_LDS
typedef int wf_v4i __attribute__((__vector_size__(16)));
// Cross-AS casts via uintptr_t round-trip (flat->LDS = low-32-bit truncation, flat->global = identity).
#define WF_GAS(p) ((__attribute__((address_space(1))) wf_v4i*)(uintptr_t)(p))
#define WF_LAS(p) ((__attribute__((address_space(3))) wf_v4i*)(uintptr_t)(p))
#endif

__global__ __launch_bounds__(TPB) void wiener_rangespace_kernel(
    const float* __restrict__ es, const float* __restrict__ en,
    const float* __restrict__ mx, float* __restrict__ out)
{
    const int tid  = threadIdx.x;
    const int bin0 = blockIdx.x * TPB;
    const int bin  = bin0 + tid;

    float v_s[16], v_n[16], v_x[16];

#if WF_ASYNC_LDS
    // ------------- CDNA5 async tensor-data path: stage (s,n,x) tiles into LDS -------------
    __shared__ float smem[3 * TPB * kBinStrideLds];   // 60 KB
    {
        const bool fullTile = (bin0 + TPB) <= kBins;  // block-uniform -> scalar branch
        if (fullTile) {
            #pragma unroll
            for (int arr = 0; arr < 3; ++arr) {
                const float* g = (arr == 0) ? es : (arr == 1) ? en : mx;
                float* lbase = &smem[arr * (TPB * kBinStrideLds)];
                #pragma unroll
                for (int j = 0; j < 4; ++j) {
                    const int  loc = j * TPB + tid;            // float4 chunk within tile
                    const long gf4 = (long)bin0 * 4 + loc;     // global float4 index (in range)
                    const float* src = g + gf4 * 4;
                    float*       dst = lbase + (loc >> 2) * kBinStrideLds + (loc & 3) * 4;
                    __builtin_amdgcn_global_load_async_to_lds_b128(WF_GAS(src), WF_LAS(dst), 0, 0);
                }
            }
        } else {
            #pragma unroll
            for (int arr = 0; arr < 3; ++arr) {
                const float* g = (arr == 0) ? es : (arr == 1) ? en : mx;
                float* lbase = &smem[arr * (TPB * kBinStrideLds)];
                #pragma unroll
                for (int j = 0; j < 4; ++j) {
                    const int  loc = j * TPB + tid;
                    const long gf4 = (long)bin0 * 4 + loc;
                    if (gf4 < (long)kBins * 4) {
                        const float* src = g + gf4 * 4;
                        float*       dst = lbase + (loc >> 2) * kBinStrideLds + (loc & 3) * 4;
                        __builtin_amdgcn_global_load_async_to_lds_b128(WF_GAS(src), WF_LAS(dst), 0, 0);
                    }
                }
            }
        }
        __builtin_amdgcn_s_wait_asynccnt(0);
        __syncthreads();
    }
    if (bin >= kBins) return;
    #pragma unroll
    for (int j = 0; j < 4; ++j) {
        *(float4*)&v_s[j * 4] = *(const float4*)&smem[0 * (TPB * kBinStrideLds) + tid * kBinStrideLds + j * 4];
        *(float4*)&v_n[j * 4] = *(const float4*)&smem[1 * (TPB * kBinStrideLds) + tid * kBinStrideLds + j * 4];
        *(float4*)&v_x[j * 4] = *(const float4*)&smem[2 * (TPB * kBinStrideLds) + tid * kBinStrideLds + j * 4];
    }
#else
    // ------------- direct streaming fallback (full cacheline utilization) -------------
    if (bin >= kBins) return;
    #pragma unroll
    for (int j = 0; j < 4; ++j) {
        *(float4*)&v_s[j * 4] = ((const float4*)es)[(long)bin * 4 + j];
        *(float4*)&v_n[j * 4] = ((const float4*)en)[(long)bin * 4 + j];
        *(float4*)&v_x[j * 4] = ((const float4*)mx)[(long)bin * 4 + j];
    }
#endif

    // ---- 5 complex dot products over M=8: a = s^H s, c = n^H n, d = s^H n, p = s^H x, q = n^H x
    float a = 0.f, c = 0.f;
    float dre = 0.f, dim = 0.f, pre = 0.f, pim = 0.f, qre = 0.f, qim = 0.f;
    #pragma unroll
    for (int m = 0; m < kM; ++m) {
        const float sr = v_s[2 * m], si = v_s[2 * m + 1];
        const float nr = v_n[2 * m], ni = v_n[2 * m + 1];
        const float xr = v_x[2 * m], xi = v_x[2 * m + 1];
        a   = fmaf(sr, sr, fmaf(si, si, a));
        c   = fmaf(nr, nr, fmaf(ni, ni, c));
        dre = fmaf(sr, nr, fmaf(si, ni, dre));
        dim = fmaf(sr, ni, fmaf(-si, nr, dim));
        pre = fmaf(sr, xr, fmaf(si, xi, pre));
        pim = fmaf(sr, xi, fmaf(-si, xr, pim));
        qre = fmaf(nr, xr, fmaf(ni, xi, qre));
        qim = fmaf(nr, xi, fmaf(-ni, xr, qim));
    }

    // ---- fp64 2x2 tail (handles rare near-parallel s,n cancellation in det); ~25 ops, free under BW roof
    const double g11 = (double)a + (double)WF_EPS;
    const double g22 = (double)c + (double)WF_EPS;
    const double Dre = dre, Dim = dim;
    const double det = g11 * g22 - (Dre * Dre + Dim * Dim);   // real, > 0
    const double inv = 1.0 / det;
    // t = g22 * p - d * q
    const double tre = g22 * (double)pre - (Dre * (double)qre - Dim * (double)qim);
    const double tim = g22 * (double)pim - (Dre * (double)qim + Dim * (double)qre);
    // bf = s0 * t / det
    const double s0r = (double)v_s[0], s0i = (double)v_s[1];
    float2 o;
    o.x = (float)((s0r * tre - s0i * tim) * inv);
    o.y = (float)((s0r * tim + s0i * tre) * inv);
    *(float2*)&out[2 * (long)bin] = o;
}

extern "C" void kernel_launch(void* const* d_in, const int* in_sizes, int n_in,
                              void* d_out, int out_size, void* d_ws, size_t ws_size,
                              hipStream_t stream) {
    (void)in_sizes; (void)n_in; (void)out_size; (void)d_ws; (void)ws_size;
    const float* es = (const float*)d_in[0];   // esti_s (B,T,F,M,2)
    const float* en = (const float*)d_in[1];   // esti_n
    const float* mx = (const float*)d_in[2];   // mix
    float*       out = (float*)d_out;          // (B,T,F,2)

    const int blocks = (kBins + TPB - 1) / TPB;   // 2008
    hipLaunchKernelGGL(wiener_rangespace_kernel, dim3(blocks), dim3(TPB), 0, stream,
                       es, en, mx, out);
}